// Group_84559316124286
// MI455X (gfx1250) — compile-verified
//
#include <hip/hip_runtime.h>
#include <hip/hip_bf16.h>

typedef __attribute__((ext_vector_type(2))) float v2f;
typedef __attribute__((ext_vector_type(8))) float v8f;

#define KNN   32     // k nearest
#define CPB   16     // centers per block (one WMMA N-tile)
#define WAVES 8
#define TPB   (WAVES * 32)
#define TPI   2      // tiles per stage (WMMAs per stage)

__device__ __forceinline__ void knn_insert(float d, int idx,
                                           float (&dl)[KNN], int (&il)[KNN]) {
  // dl[] kept sorted ascending; replace max then one bubble pass.
  if (d < dl[KNN - 1]) {
    dl[KNN - 1] = d;
    il[KNN - 1] = idx;
#pragma unroll
    for (int j = KNN - 1; j > 0; --j) {
      float a = dl[j - 1], b = dl[j];
      if (b < a) {
        dl[j - 1] = b; dl[j] = a;
        int t = il[j - 1]; il[j - 1] = il[j]; il[j] = t;
      }
    }
  }
}

// Load this lane's point row for tiles t and t+1 (wave-uniform call sites).
__device__ __forceinline__ void load_pair(const float* __restrict__ xyz,
                                          int t, int c,
                                          float& x0, float& y0, float& z0,
                                          float& x1, float& y1, float& z1) {
  const int p0 = (t << 4) + c;
  const int p1 = ((t + 1) << 4) + c;
  x0 = xyz[p0 * 3 + 0]; y0 = xyz[p0 * 3 + 1]; z0 = xyz[p0 * 3 + 2];
  x1 = xyz[p1 * 3 + 0]; y1 = xyz[p1 * 3 + 1]; z1 = xyz[p1 * 3 + 2];
}

// Two WMMAs (tiles t, t+1) + top-k insertion of the 16 resulting candidates.
__device__ __forceinline__ void process_pair(int t, bool hi, v2f bfrag,
                                             float x0, float y0, float z0,
                                             float x1, float y1, float z1,
                                             float (&dl)[KNN], int (&il)[KNN]) {
  // A fragments: points, M x K = 16 x 4
  // lanes 0-15 : v0 = K0 = px, v1 = K1 = py
  // lanes 16-31: v0 = K2 = pz, v1 = K3 = ||p||^2
  v2f af0, af1;
  af0.x = hi ? z0 : x0;
  af0.y = hi ? (x0 * x0 + y0 * y0 + z0 * z0) : y0;
  af1.x = hi ? z1 : x1;
  af1.y = hi ? (x1 * x1 + y1 * y1 + z1 * z1) : y1;

  v8f acc0 = {0.f, 0.f, 0.f, 0.f, 0.f, 0.f, 0.f, 0.f};
  v8f acc1 = {0.f, 0.f, 0.f, 0.f, 0.f, 0.f, 0.f, 0.f};
  acc0 = __builtin_amdgcn_wmma_f32_16x16x4_f32(
      false, af0, false, bfrag, (short)0, acc0, false, false);
  acc1 = __builtin_amdgcn_wmma_f32_16x16x4_f32(
      false, af1, false, bfrag, (short)0, acc1, false, false);

  // acc[r] = ||p_m||^2 - 2 p_m . c  for m = tile*16 + r (+8 on upper half)
  const int base0 = (t << 4) + (hi ? 8 : 0);
  const int base1 = ((t + 1) << 4) + (hi ? 8 : 0);
#pragma unroll
  for (int r = 0; r < 8; ++r) knn_insert(acc0[r], base0 + r, dl, il);
#pragma unroll
  for (int r = 0; r < 8; ++r) knn_insert(acc1[r], base1 + r, dl, il);
}

// One block handles 16 centers. Each wave scans strided pairs of 16-point
// tiles; each tile is one v_wmma_f32_16x16x4_f32 producing a
// 16(points) x 16(centers) tile of (||p||^2 - 2 p.c). In the C/D layout each
// lane's 8 accumulators all belong to center (lane & 15), so the top-k lists
// live entirely in registers. Point loads are double-buffered across two
// unrolled stages; all loads are unconditional (last prefetch index is
// clamped branchlessly) so the waitcnt pass can use relaxed load counts and
// the insert chains fully hide L2 latency.
__global__ __launch_bounds__(TPB) void knn_group_kernel(
    const float* __restrict__ xyz,      // (N,3)
    const int*   __restrict__ sample,   // (G,)
    float*       __restrict__ out,      // concat outputs, all as float
    int N, int G) {
  __shared__ float sD[CPB * 16 * KNN];  // per-center 16 sorted sublists
  __shared__ int   sI[CPB * 16 * KNN];

  const int tid  = threadIdx.x;
  const int lane = tid & 31;
  const int wave = tid >> 5;
  const int c    = lane & 15;          // this lane's center column
  const bool hi  = lane >= 16;

  // ---- B fragment: centers, K x N = 4 x 16 -------------------------------
  // lanes 0-15 : v0 = K0 = -2*cx, v1 = K1 = -2*cy
  // lanes 16-31: v0 = K2 = -2*cz, v1 = K3 = 1.0
  const int g0 = blockIdx.x * CPB;
  int   cs = sample[g0 + c];
  float cx = xyz[cs * 3 + 0];
  float cy = xyz[cs * 3 + 1];
  float cz = xyz[cs * 3 + 2];
  cx = (cx == cx) ? cx : 0.0f;   // NaN -> 0 per reference
  cy = (cy == cy) ? cy : 0.0f;
  cz = (cz == cz) ? cz : 0.0f;
  v2f bfrag;
  bfrag.x = hi ? (-2.0f * cz) : (-2.0f * cx);
  bfrag.y = hi ? 1.0f         : (-2.0f * cy);

  // ---- per-lane sorted top-k (registers) ---------------------------------
  float dl[KNN];
  int   il[KNN];
#pragma unroll
  for (int j = 0; j < KNN; ++j) { dl[j] = 3.402823466e38f; il[j] = 0; }

  // ---- scan all points: two double-buffered stages per trip --------------
  // ntiles is divisible by 2*STEP (4096 / 32), so both stages are always
  // valid; only the final stage-B prefetch needs an index clamp.
  const int ntiles = N >> 4;           // 16-point tiles
  const int STEP   = WAVES * TPI;      // tiles consumed per stage (16)
  const int tfirst = wave * TPI;

  float ax0, ay0, az0, ax1, ay1, az1;  // consumed by stage A
  float bx0, by0, bz0, bx1, by1, bz1;  // consumed by stage B

  load_pair(xyz, tfirst, c, ax0, ay0, az0, ax1, ay1, az1);

  for (int t = tfirst; t < ntiles; t += 2 * STEP) {
    // ---- stage A: prefetch buffer B, process buffer A ----
    const int tB = t + STEP;                    // always < ntiles
    load_pair(xyz, tB, c, bx0, by0, bz0, bx1, by1, bz1);
    process_pair(t, hi, bfrag, ax0, ay0, az0, ax1, ay1, az1, dl, il);

    // ---- stage B: prefetch buffer A (clamped), process buffer B ----
    int tA = t + 2 * STEP;
    tA = (tA < ntiles) ? tA : tfirst;           // branchless uniform clamp
    load_pair(xyz, tA, c, ax0, ay0, az0, ax1, ay1, az1);
    process_pair(tB, hi, bfrag, bx0, by0, bz0, bx1, by1, bz1, dl, il);
  }

  // ---- dump sorted sublists to LDS ---------------------------------------
  const int slist = wave * 2 + (hi ? 1 : 0);       // 0..15
  const int off   = (c * 16 + slist) * KNN;
#pragma unroll
  for (int j = 0; j < KNN; ++j) { sD[off + j] = dl[j]; sI[off + j] = il[j]; }
  __syncthreads();

  // ---- 16-way merge of sorted lists; one thread per center ---------------
  if (tid < CPB) {
    const int gc  = g0 + tid;
    const int csi = sample[gc];
    float ccx = xyz[csi * 3 + 0];
    float ccy = xyz[csi * 3 + 1];
    float ccz = xyz[csi * 3 + 2];
    ccx = (ccx == ccx) ? ccx : 0.0f;
    ccy = (ccy == ccy) ? ccy : 0.0f;
    ccz = (ccz == ccz) ? ccz : 0.0f;

    const long NB_OFF  = 0;                              // (G,KNN,3)
    const long CEN_OFF = (long)G * KNN * 3;              // (G,3)
    const long IDX_OFF = CEN_OFF + (long)G * 3;          // (G,KNN)
    const long CID_OFF = IDX_OFF + (long)G * KNN;        // (G,)

    out[CEN_OFF + gc * 3 + 0] = ccx;
    out[CEN_OFF + gc * 3 + 1] = ccy;
    out[CEN_OFF + gc * 3 + 2] = ccz;
    out[CID_OFF + gc]         = (float)csi;

    int head[16];
#pragma unroll
    for (int s = 0; s < 16; ++s) head[s] = 0;

    for (int j = 0; j < KNN; ++j) {
      float best = __builtin_huge_valf();
      int   bs   = 0;
#pragma unroll
      for (int s = 0; s < 16; ++s) {
        const int h = head[s];
        const float v = (h < KNN) ? sD[(tid * 16 + s) * KNN + h]
                                  : __builtin_huge_valf();
        if (v < best) { best = v; bs = s; }
      }
      const int idx = sI[(tid * 16 + bs) * KNN + head[bs]];
      head[bs]++;

      const long nb = NB_OFF + ((long)gc * KNN + j) * 3;
      out[nb + 0] = xyz[idx * 3 + 0] - ccx;
      out[nb + 1] = xyz[idx * 3 + 1] - ccy;
      out[nb + 2] = xyz[idx * 3 + 2] - ccz;
      out[IDX_OFF + (long)gc * KNN + j] = (float)idx;
    }
  }
}

extern "C" void kernel_launch(void* const* d_in, const int* in_sizes, int n_in,
                              void* d_out, int out_size, void* d_ws, size_t ws_size,
                              hipStream_t stream) {
  const float* xyz    = (const float*)d_in[0];
  const int*   sample = (const int*)d_in[1];
  float*       out    = (float*)d_out;

  const int N = in_sizes[0] / 3;   // 65536
  const int G = in_sizes[1];       // 1024

  const int blocks = G / CPB;      // 64
  knn_group_kernel<<<blocks, TPB, 0, stream>>>(xyz, sample, out, N, G);
}